// GreedyMatcher_43061342110271
// MI455X (gfx1250) — compile-verified
//
#include <hip/hip_runtime.h>
#include <hip/hip_bf16.h>

// Batched log-barrier LP interior point for GreedyMatcher.
//   M=128 constraints, N=512 variables, BATCH=32, OUTER=10, INNER=2, T0=1, MU=4.
// One workgroup (256 threads / 8 wave32) per batch element.
// Hessian H = S^T diag(1/r^2) S + diag(1/x^2) formed with V_WMMA_F32_16X16X4_F32
// in d_ws (1MB per block -> 32MB ws total); blocked (NB=16) Cholesky whose trailing
// updates also run on V_WMMA_F32_16X16X4_F32; blocked triangular solves (NB=16).

#define LP_M     128
#define LP_N     512
#define LP_BATCH 32
#define LP_OUTER 10
#define LP_INNER 2
#define NTHREADS 256
#define NBLK     32          // LP_N / 16

typedef __attribute__((ext_vector_type(2))) float v2f;
typedef __attribute__((ext_vector_type(8))) float v8f;

__device__ __forceinline__ float block_min(float v, volatile float* red, int tid) {
    red[tid] = v;
    __syncthreads();
    for (int off = NTHREADS / 2; off > 0; off >>= 1) {
        if (tid < off) red[tid] = fminf((float)red[tid], (float)red[tid + off]);
        __syncthreads();
    }
    float r = red[0];
    __syncthreads();
    return r;
}

__device__ __forceinline__ float block_max(float v, volatile float* red, int tid) {
    red[tid] = v;
    __syncthreads();
    for (int off = NTHREADS / 2; off > 0; off >>= 1) {
        if (tid < off) red[tid] = fmaxf((float)red[tid], (float)red[tid + off]);
        __syncthreads();
    }
    float r = red[0];
    __syncthreads();
    return r;
}

__global__ void __launch_bounds__(NTHREADS)
lp_barrier_kernel(const float* __restrict__ X,   // [BATCH, 128]  (b vectors)
                  const float* __restrict__ S,   // [128, 512]
                  const float* __restrict__ W,   // [512]
                  float* __restrict__ out,       // [BATCH, 512]
                  float* __restrict__ ws)        // >= BATCH * 512 * 512 floats
{
    const int tid  = threadIdx.x;
    const int lane = tid & 31;
    const int wave = tid >> 5;          // 0..7
    const int bidx = blockIdx.x;        // batch element
    const int m    = lane & 15;         // fragment M/N index
    const int half = lane >> 4;         // fragment K-group select

    __shared__ float xs[LP_N];          // x
    __shared__ float rhs[LP_N];         // -g, then y, then dx (in place)
    __shared__ float invxS[LP_N];       // 1/x
    __shared__ float bS[LP_M];          // b
    __shared__ float rS[LP_M];          // r = b - Sx
    __shared__ float invrS[LP_M];       // 1/r
    __shared__ float red[NTHREADS];     // reduction scratch
    __shared__ float ld16[256];         // 16x16 diagonal block scratch

    float* H = ws + (size_t)bidx * LP_N * LP_N;

    // ---------------- init: x0 = 0.5 * min(b) / max_row_sum(S) ----------------
    for (int k = tid; k < LP_M; k += NTHREADS) bS[k] = X[bidx * LP_M + k];

    float myMax = -3.4e38f;
    if (tid < LP_M) {
        float s = 0.f;
        const float* Srow = S + tid * LP_N;
        for (int j = 0; j < LP_N; ++j) s += Srow[j];
        myMax = s;
    }
    float maxRowSum = block_max(myMax, red, tid);   // syncs inside (bS now visible)

    float myMin = 3.4e38f;
    for (int k = tid; k < LP_M; k += NTHREADS) myMin = fminf(myMin, bS[k]);
    float minB = block_min(myMin, red, tid);

    float x0 = 0.5f * minB / maxRowSum;
    for (int j = tid; j < LP_N; j += NTHREADS) xs[j] = x0;
    __syncthreads();

    // ---------------- barrier / Newton loop ----------------
    float tbar = 1.0f;                       // T0
    for (int outer = 0; outer < LP_OUTER; ++outer) {
        for (int inner = 0; inner < LP_INNER; ++inner) {
            // -- slack r = b - S x ;  1/r --
            if (tid < LP_M) {
                float acc = bS[tid];
                const float* Srow = S + tid * LP_N;
                for (int j = 0; j < LP_N; ++j) acc -= Srow[j] * xs[j];
                rS[tid]    = acc;
                invrS[tid] = 1.0f / acc;
            }
            __syncthreads();

            // -- rhs = -g = t*w + 1/x - S^T (1/r) ;  invx --
            for (int j = tid; j < LP_N; j += NTHREADS) {
                float ix = 1.0f / xs[j];
                invxS[j] = ix;
                float dot = 0.f;
                for (int k = 0; k < LP_M; ++k) dot += S[k * LP_N + j] * invrS[k];
                rhs[j] = tbar * W[j] + ix - dot;
            }
            __syncthreads();

            // -- H = A^T A,  A = diag(1/r) S  (f32 WMMA 16x16x4, upper tiles + mirror) --
            for (int t = wave; t < 528; t += 8) {       // 528 = 32*33/2, uniform per wave
                int rem = t, I = 0;                     // triangular index -> (I, J), J >= I
                while (rem >= (NBLK - I)) { rem -= (NBLK - I); ++I; }
                int J  = I + rem;
                int i0 = I << 4, j0 = J << 4;

                v8f acc = {};
                #pragma unroll
                for (int kk = 0; kk < LP_M; kk += 4) {
                    int ka = kk + (half << 1);          // K for vgpr .x ; .y is ka+1
                    float ir0 = invrS[ka];
                    float ir1 = invrS[ka + 1];
                    v2f a, b;
                    a.x = ir0 * S[ka * LP_N + i0 + m];
                    a.y = ir1 * S[(ka + 1) * LP_N + i0 + m];
                    b.x = ir0 * S[ka * LP_N + j0 + m];
                    b.y = ir1 * S[(ka + 1) * LP_N + j0 + m];
                    acc = __builtin_amdgcn_wmma_f32_16x16x4_f32(
                        false, a, false, b, (short)0, acc, false, false);
                }
                // C/D layout: vgpr v holds M=v (lanes 0-15) / M=v+8 (lanes 16-31), N=lane&15
                #pragma unroll
                for (int v = 0; v < 8; ++v) {
                    int row = i0 + v + (half << 3);
                    int col = j0 + m;
                    float val = acc[v];
                    H[row * LP_N + col] = val;
                    H[col * LP_N + row] = val;          // mirror (fills full H)
                }
            }
            __syncthreads();
            for (int i = tid; i < LP_N; i += NTHREADS)
                H[i * LP_N + i] += invxS[i] * invxS[i];
            __syncthreads();

            // -- blocked Cholesky, NB=16; WMMA trailing updates; lower triangle only --
            for (int p = 0; p < NBLK; ++p) {
                const int j0 = p << 4;

                // 1) stage diagonal block into LDS
                ld16[tid] = H[(j0 + (tid >> 4)) * LP_N + j0 + (tid & 15)];
                __syncthreads();

                // 2) 16x16 left-looking Cholesky (single thread; ~700 flop)
                if (tid == 0) {
                    for (int c = 0; c < 16; ++c) {
                        float d = ld16[c * 16 + c];
                        for (int cc = 0; cc < c; ++cc) d -= ld16[c * 16 + cc] * ld16[c * 16 + cc];
                        d = sqrtf(d);
                        ld16[c * 16 + c] = d;
                        for (int r2 = c + 1; r2 < 16; ++r2) {
                            float v = ld16[r2 * 16 + c];
                            for (int cc = 0; cc < c; ++cc) v -= ld16[r2 * 16 + cc] * ld16[c * 16 + cc];
                            ld16[r2 * 16 + c] = v / d;
                        }
                    }
                }
                __syncthreads();

                // 3) write back L_D (lower triangle of diagonal block)
                {
                    int rr = tid >> 4, cc = tid & 15;
                    if (cc <= rr) H[(j0 + rr) * LP_N + j0 + cc] = ld16[tid];
                }
                __syncthreads();

                // 4) panel solve: L21 = A21 * L_D^{-T}, one row per thread, register-resident
                for (int r = j0 + 16 + tid; r < LP_N; r += NTHREADS) {
                    float lr[16];
                    #pragma unroll
                    for (int c = 0; c < 16; ++c) {
                        float v = H[r * LP_N + j0 + c];
                        #pragma unroll
                        for (int cc = 0; cc < c; ++cc) v -= lr[cc] * ld16[c * 16 + cc];
                        lr[c] = v / ld16[c * 16 + c];
                    }
                    #pragma unroll
                    for (int c = 0; c < 16; ++c) H[r * LP_N + j0 + c] = lr[c];
                }
                __syncthreads();

                // 5) WMMA trailing update: C(BI,BJ) -= P_BI * P_BJ^T, lower tiles only
                {
                    const int T  = NBLK - 1 - p;        // trailing block count per dim
                    const int nt = T * (T + 1) / 2;
                    for (int t = wave; t < nt; t += 8) {
                        int rem = t, Ib = 0;            // lower-triangle index -> (Ib, Jb<=Ib)
                        while (rem > Ib) { rem -= Ib + 1; ++Ib; }
                        const int i0  = (p + 1 + Ib) << 4;
                        const int j0t = (p + 1 + rem) << 4;

                        v8f acc;
                        #pragma unroll
                        for (int v = 0; v < 8; ++v)
                            acc[v] = H[(i0 + v + (half << 3)) * LP_N + j0t + m];
                        #pragma unroll
                        for (int kk = 0; kk < 16; kk += 4) {
                            int ka = kk + (half << 1);
                            v2f a = *(const v2f*)&H[(i0 + m) * LP_N + j0 + ka];
                            a = -a;   // f32 WMMA has no A/B-neg (NEG=CNeg only): negate in VALU
                            v2f b = *(const v2f*)&H[(j0t + m) * LP_N + j0 + ka];
                            acc = __builtin_amdgcn_wmma_f32_16x16x4_f32(
                                false, a, false, b, (short)0, acc, false, false);
                        }
                        #pragma unroll
                        for (int v = 0; v < 8; ++v)
                            H[(i0 + v + (half << 3)) * LP_N + j0t + m] = acc[v];
                    }
                }
                __syncthreads();
            }

            // -- blocked forward solve: L y = rhs (in place), 2 syncs per 16-panel --
            for (int p = 0; p < NBLK; ++p) {
                const int j0 = p << 4;
                if (tid == 0) {
                    #pragma unroll
                    for (int c = 0; c < 16; ++c) {
                        float v = rhs[j0 + c];
                        for (int cc = 0; cc < c; ++cc)
                            v -= H[(j0 + c) * LP_N + j0 + cc] * rhs[j0 + cc];
                        rhs[j0 + c] = v / H[(j0 + c) * LP_N + j0 + c];
                    }
                }
                __syncthreads();
                for (int i = j0 + 16 + tid; i < LP_N; i += NTHREADS) {
                    float acc = rhs[i];
                    const float* Lr = &H[i * LP_N + j0];   // contiguous 16-float row segment
                    #pragma unroll
                    for (int c = 0; c < 16; ++c) acc -= Lr[c] * rhs[j0 + c];
                    rhs[i] = acc;
                }
                __syncthreads();
            }

            // -- blocked backward solve: L^T dx = y (in place; rhs becomes dx) --
            for (int p = NBLK - 1; p >= 0; --p) {
                const int j0 = p << 4;
                if (tid == 0) {
                    #pragma unroll
                    for (int c = 15; c >= 0; --c) {
                        float v = rhs[j0 + c];
                        for (int cc = c + 1; cc < 16; ++cc)
                            v -= H[(j0 + cc) * LP_N + j0 + c] * rhs[j0 + cc];
                        rhs[j0 + c] = v / H[(j0 + c) * LP_N + j0 + c];
                    }
                }
                __syncthreads();
                for (int i = tid; i < j0; i += NTHREADS) {
                    float acc = rhs[i];
                    #pragma unroll
                    for (int c = 0; c < 16; ++c)
                        acc -= H[(j0 + c) * LP_N + i] * rhs[j0 + c];   // L^T[i][.] = L[.][i]
                    rhs[i] = acc;
                }
                __syncthreads();
            }

            // -- fraction-to-boundary step --
            float amin = 3.4e38f;
            if (tid < LP_M) {
                float sdx = 0.f;
                const float* Srow = S + tid * LP_N;
                for (int j = 0; j < LP_N; ++j) sdx += Srow[j] * rhs[j];
                if (sdx > 0.f) amin = rS[tid] / sdx;
            }
            for (int j = tid; j < LP_N; j += NTHREADS) {
                float d = rhs[j];
                if (d < 0.f) amin = fminf(amin, -xs[j] / d);
            }
            float am = block_min(amin, red, tid);
            float a  = fminf(1.0f, 0.95f * am);
            for (int j = tid; j < LP_N; j += NTHREADS) xs[j] += a * rhs[j];
            __syncthreads();
        }
        tbar *= 4.0f;    // MU
    }

    // ---------------- output ----------------
    for (int j = tid; j < LP_N; j += NTHREADS)
        out[bidx * LP_N + j] = xs[j];
}

extern "C" void kernel_launch(void* const* d_in, const int* in_sizes, int n_in,
                              void* d_out, int out_size, void* d_ws, size_t ws_size,
                              hipStream_t stream) {
    // inputs (setup_inputs order): X [32,8,16] f32, batch_size (scalar), S [128,512] f32, W [512] f32
    const float* X = (const float*)d_in[0];
    const float* S = (const float*)d_in[2];
    const float* W = (const float*)d_in[3];
    float* out = (float*)d_out;
    float* ws  = (float*)d_ws;          // needs >= 32 * 512 * 512 * 4 = 32 MB

    (void)in_sizes; (void)n_in; (void)out_size; (void)ws_size;

    lp_barrier_kernel<<<dim3(LP_BATCH), dim3(NTHREADS), 0, stream>>>(X, S, W, out, ws);
}